// GraphConv_39084202394051
// MI455X (gfx1250) — compile-verified
//
#include <hip/hip_runtime.h>

typedef float v2f __attribute__((ext_vector_type(2)));
typedef float v4f __attribute__((ext_vector_type(4)));
typedef float v8f __attribute__((ext_vector_type(8)));

#define N_NODES 50000
#define D 256
#define KNB 16
#define KT 32          // k-tile staged in LDS per iteration
#define PITCH 36       // LDS row pitch in floats: 16B-aligned, bank-conflict-free

// ---------------------------------------------------------------------------
// Kernel 1: x = relu(feats @ W^T + b)   [N, 256]
// One block (8 wave32s) computes a 32-row x 256-col strip.
// Each wave: 2 m-tiles x 2 n-tiles (2x2 register blocking) -> 4 WMMA per
// k-step off 4 fragment loads (A and B fragments each reused twice).
// ---------------------------------------------------------------------------
__global__ __launch_bounds__(256) void gemm_bias_relu(
    const float* __restrict__ feats,
    const float* __restrict__ W,      // [D][D], row n holds W[n, :]
    const float* __restrict__ bias,   // [D]
    float* __restrict__ x)            // [N][D]
{
    __shared__ float sW[D * PITCH];   // W[n][k0..k0+31]  (n-major, pitch 36)
    __shared__ float sA[32 * PITCH];  // feats[m0..m0+31][k0..k0+31]

    const int tid  = threadIdx.x;
    const int wave = tid >> 5;
    const int lane = tid & 31;
    const int m0   = blockIdx.x * 32;
    const int n0   = wave * 32;       // this wave covers columns n0 .. n0+31

    v8f acc00 = {};                   // (m-tile 0, n-tile n0)
    v8f acc01 = {};                   // (m-tile 0, n-tile n0+16)
    v8f acc10 = {};                   // (m-tile 1, n-tile n0)
    v8f acc11 = {};                   // (m-tile 1, n-tile n0+16)

    const int row = lane & 15;        // M (for A) / N (for B) within a tile
    const int kh  = (lane >> 4) * 2;  // K sub-offset: lanes 0-15 -> K0/K1, 16-31 -> K2/K3

    for (int k0 = 0; k0 < D; k0 += KT) {
        // ---- stage W[:, k0:k0+32] into LDS: 256 rows x 32 floats ----
        {
            const int ki = (tid & 7) * 4;       // 0,4,...,28
            int n = tid >> 3;                   // 0..31
            #pragma unroll
            for (int p = 0; p < 8; ++p, n += 32) {
                v4f w4 = *(const v4f*)(W + n * D + k0 + ki);
                *(v4f*)(sW + n * PITCH + ki) = w4;   // ds_store_b128
            }
        }
        // ---- stage feats[m0:m0+32, k0:k0+32] (clamped for last block) ----
        {
            const int m  = tid >> 3;            // 0..31
            const int ki = (tid & 7) * 4;
            int srcRow = m0 + m;
            if (srcRow > N_NODES - 1) srcRow = N_NODES - 1;
            v4f a4 = *(const v4f*)(feats + srcRow * D + k0 + ki);
            *(v4f*)(sA + m * PITCH + ki) = a4;
        }
        __syncthreads();

        // ---- 8 k-steps of 4: 32 WMMAs per k-tile per wave ----
        #pragma unroll
        for (int kk = 0; kk < KT; kk += 4) {
            v2f a0, a1, b0, b1;
            // A 16x4 f32 fragments (ISA 7.12.2): lane=M (mod 16), vgpr0/1 = K pair
            a0.x = sA[row * PITCH + kk + kh];
            a0.y = sA[row * PITCH + kk + kh + 1];
            a1.x = sA[(16 + row) * PITCH + kk + kh];
            a1.y = sA[(16 + row) * PITCH + kk + kh + 1];
            // B 4x16 fragments = W^T slice: B[k][n] = W[n][k]
            b0.x = sW[(n0 + row) * PITCH + kk + kh];
            b0.y = sW[(n0 + row) * PITCH + kk + kh + 1];
            b1.x = sW[(n0 + 16 + row) * PITCH + kk + kh];
            b1.y = sW[(n0 + 16 + row) * PITCH + kk + kh + 1];
            // 8 args: (neg_a, A, neg_b, B, c_mod, C, reuse_a, reuse_b)
            acc00 = __builtin_amdgcn_wmma_f32_16x16x4_f32(
                false, a0, false, b0, (short)0, acc00, false, false);
            acc01 = __builtin_amdgcn_wmma_f32_16x16x4_f32(
                false, a0, false, b1, (short)0, acc01, false, false);
            acc10 = __builtin_amdgcn_wmma_f32_16x16x4_f32(
                false, a1, false, b0, (short)0, acc10, false, false);
            acc11 = __builtin_amdgcn_wmma_f32_16x16x4_f32(
                false, a1, false, b1, (short)0, acc11, false, false);
        }
        __syncthreads();
    }

    // ---- epilogue: bias + relu + coalesced store ----
    // C/D layout: vgpr v, lane l -> m = v + 8*(l/16), n = l%16
    const int mh  = (lane >> 4) * 8;
    const float bv0 = bias[n0 + row];
    const float bv1 = bias[n0 + 16 + row];
    #pragma unroll
    for (int v = 0; v < 8; ++v) {
        const int mA = m0 + v + mh;        // m-tile 0
        const int mB = m0 + 16 + v + mh;   // m-tile 1
        if (mA < N_NODES) {
            x[mA * D + n0 + row]      = fmaxf(acc00[v] + bv0, 0.0f);
            x[mA * D + n0 + 16 + row] = fmaxf(acc01[v] + bv1, 0.0f);
        }
        if (mB < N_NODES) {
            x[mB * D + n0 + row]      = fmaxf(acc10[v] + bv0, 0.0f);
            x[mB * D + n0 + 16 + row] = fmaxf(acc11[v] + bv1, 0.0f);
        }
    }
}

// ---------------------------------------------------------------------------
// Kernel 2: pooled[i] = mean_j x[edge[i,j]]   (gather out of L2-resident x)
// 4 nodes per 256-thread block; 64 lanes x float4 per node row.
// ---------------------------------------------------------------------------
__global__ __launch_bounds__(256) void pool_mean(
    const float* __restrict__ x,
    const int* __restrict__ edges,    // [N][16]
    float* __restrict__ out)          // [N][D]
{
    const int node = blockIdx.x * 4 + (threadIdx.x >> 6);
    const int sub  = threadIdx.x & 63;          // column group: 4 floats each
    const int* e = edges + node * KNB;
    v4f acc = {};
    #pragma unroll
    for (int j = 0; j < KNB; ++j) {
        const int nb = e[j];
        v4f v = *(const v4f*)(x + nb * D + sub * 4);  // global_load_b128
        acc += v;
    }
    acc *= 0.0625f;                                   // 1/16
    *(v4f*)(out + node * D + sub * 4) = acc;
}

// ---------------------------------------------------------------------------
extern "C" void kernel_launch(void* const* d_in, const int* in_sizes, int n_in,
                              void* d_out, int out_size, void* d_ws, size_t ws_size,
                              hipStream_t stream) {
    // setup_inputs order: idx, feats, edge_dict, sadj, epoch, W, b
    const float* feats = (const float*)d_in[1];
    const int*   edges = (const int*)  d_in[2];
    const float* W     = (const float*)d_in[5];
    const float* b     = (const float*)d_in[6];
    float* x   = (float*)d_ws;          // intermediate [N][D] = 51.2 MB
    float* out = (float*)d_out;

    gemm_bias_relu<<<(N_NODES + 31) / 32, 256, 0, stream>>>(feats, W, b, x);
    pool_mean<<<N_NODES / 4, 256, 0, stream>>>(x, edges, out);
}